// GnBlock_72138270704061
// MI455X (gfx1250) — compile-verified
//
#include <hip/hip_runtime.h>

#define H  128
#define TM 64
#define TK 64
#define BSTRIDE (2 * H + 32)   // k2-row stride (floats): +32 shifts half-wave onto other banks

typedef float v2f __attribute__((ext_vector_type(2)));
typedef float v8f __attribute__((ext_vector_type(8)));

// Generic fused GEMM: out0[M,128] = (relu_in? relu(A + bin) : A) @ W[K,128] + bout (+resid)
// A rows are a concat of up to three 128-wide sources, each optionally gathered.
__global__ __launch_bounds__(256)
void wmma_gemm_kernel(const float* __restrict__ S0,
                      const float* __restrict__ S1,
                      const float* __restrict__ S2,
                      const int* __restrict__ I0,
                      const int* __restrict__ I1,
                      const float* __restrict__ W,
                      const float* __restrict__ bin,    // len K, relu(a+bin) if non-null
                      const float* __restrict__ bout,   // len 128 or null
                      const float* __restrict__ resid,  // [M,128] or null
                      float* __restrict__ out0,
                      float* __restrict__ out1,         // raw (pre-residual) copy or null
                      int M, int K)
{
  __shared__ float As[TM][TK + 4];          // padded to dodge bank conflicts
  __shared__ float Bs[TK / 2][BSTRIDE];     // K-pair interleaved: Bs[k>>1][col*2 + (k&1)]

  const int tid  = threadIdx.x;
  const int wave = tid >> 5;
  const int lane = tid & 31;
  const int m0   = blockIdx.x * TM;
  const int wm   = (wave >> 1) << 4;   // 0,16,32,48 (row tile in block)
  const int wn   = (wave & 1) << 6;    // 0,64       (col half)
  const int lm   = lane & 15;
  const int hk   = (lane >> 4) << 1;   // 0 or 2 (K sub-pair per half-wave)

  v8f acc[4] = {};

  const int nchunks = K >> 6;
  for (int kc = 0; kc < nchunks; ++kc) {
    const int si = kc >> 1;
    const float* src = (si == 0) ? S0 : (si == 1) ? S1 : S2;
    const int*   idx = (si == 0) ? I0 : (si == 1) ? I1 : nullptr;
    const int coff = (kc & 1) << 6;

    // stage A tile: 64 rows x 64 cols, gathered + optional bias/relu
    #pragma unroll
    for (int i = 0; i < 4; ++i) {
      int flat = tid + (i << 8);
      int r = flat >> 4;
      int c = (flat & 15) << 2;
      int row = m0 + r;
      float4 v = make_float4(0.f, 0.f, 0.f, 0.f);
      if (row < M) {
        int g = idx ? idx[row] : row;
        v = *(const float4*)(src + (size_t)g * H + coff + c);
      }
      if (bin) {
        const float* bp = bin + (kc << 6) + c;
        v.x = fmaxf(v.x + bp[0], 0.f);
        v.y = fmaxf(v.y + bp[1], 0.f);
        v.z = fmaxf(v.z + bp[2], 0.f);
        v.w = fmaxf(v.w + bp[3], 0.f);
      }
      *(float4*)&As[r][c] = v;
    }
    // stage W chunk 64x128, interleaving consecutive K rows so a B fragment
    // (K=k..k+1 at one column) is a single aligned 8-byte LDS load.
    #pragma unroll
    for (int i = 0; i < 4; ++i) {
      int flat = tid + (i << 8);          // 0..1023
      int r2 = flat >> 5;                 // 0..31 (pair of K rows)
      int c = (flat & 31) << 2;           // 0..124
      const float* w0 = W + (size_t)((kc << 6) + (r2 << 1)) * H + c;
      float4 a4 = *(const float4*)w0;
      float4 b4 = *(const float4*)(w0 + H);
      float* bs = &Bs[r2][c << 1];
      *(v2f*)(bs + 0) = (v2f){a4.x, b4.x};
      *(v2f*)(bs + 2) = (v2f){a4.y, b4.y};
      *(v2f*)(bs + 4) = (v2f){a4.z, b4.z};
      *(v2f*)(bs + 6) = (v2f){a4.w, b4.w};
    }
    __syncthreads();

    #pragma unroll 8
    for (int k = 0; k < TK; k += 4) {
      // A 16x4 fp32 frag: lanes 0-15 hold K0/K1, lanes 16-31 hold K2/K3
      v2f a = *(const v2f*)&As[wm + lm][k + hk];
      const int k2 = (k + hk) >> 1;
      #pragma unroll
      for (int t = 0; t < 4; ++t) {
        // B 4x16 frag: lanes index N, VGPR pair indexes K — one b64 LDS load
        v2f b = *(const v2f*)&Bs[k2][(wn + (t << 4) + lm) << 1];
        acc[t] = __builtin_amdgcn_wmma_f32_16x16x4_f32(
            false, a, false, b, (short)0, acc[t], false, false);
      }
    }
    __syncthreads();
  }

  // C/D layout: VGPR r -> row (wm+r) for lanes 0-15, (wm+8+r) for lanes 16-31
  const int rbase = m0 + wm + ((lane >> 4) << 3);
  #pragma unroll
  for (int t = 0; t < 4; ++t) {
    const int col = wn + (t << 4) + lm;
    const float bo = bout ? bout[col] : 0.f;
    #pragma unroll
    for (int r = 0; r < 8; ++r) {
      int row = rbase + r;
      if (row < M) {
        float v = acc[t][r] + bo;
        if (out1)  out1[(size_t)row * H + col] = v;
        if (resid) v += resid[(size_t)row * H + col];
        out0[(size_t)row * H + col] = v;
      }
    }
  }
}

__global__ void deg_kernel(const int* __restrict__ rcv, int* __restrict__ cnt, int E) {
  int k = blockIdx.x * blockDim.x + threadIdx.x;
  if (k < E) atomicAdd(&cnt[rcv[k]], 1);
}

__global__ void dinv_kernel(const int* __restrict__ cnt, float* __restrict__ dinv, int N) {
  int i = blockIdx.x * blockDim.x + threadIdx.x;
  if (i < N) dinv[i] = rsqrtf(1.0f + (float)cnt[i]);   // self loop -> deg >= 1
}

// out[j] = dinv_e(j)^2 * xw[j], with dinv_e(j) = (j < Nlim ? dinv[j] : 1)
__global__ void self_scale_kernel(const float* __restrict__ xw, float* __restrict__ out,
                                  const float* __restrict__ dinv, int M, int Nlim) {
  int t = blockIdx.x * blockDim.x + threadIdx.x;
  int row = t >> 5;
  if (row >= M) return;
  int c = (t & 31) << 2;
  float d = (row < Nlim) ? dinv[row] : 1.0f;
  float s = d * d;
  float4 v = *(const float4*)(xw + (size_t)row * H + c);
  v.x *= s; v.y *= s; v.z *= s; v.w *= s;
  *(float4*)(out + (size_t)row * H + c) = v;
}

// dst[rcv[k]] += dinv[snd[k]]*dinv[rcv[k]] * src[snd[k]]
__global__ void scatter_norm_kernel(const float* __restrict__ src, float* __restrict__ dst,
                                    const int* __restrict__ snd, const int* __restrict__ rcv,
                                    const float* __restrict__ dinv, int E) {
  int t = blockIdx.x * blockDim.x + threadIdx.x;
  int k = t >> 5;
  if (k >= E) return;
  int c = (t & 31) << 2;
  int s = snd[k], r = rcv[k];
  float w = dinv[s] * dinv[r];
  float4 v = *(const float4*)(src + (size_t)s * H + c);
  float* d = dst + (size_t)r * H + c;
  atomicAdd(d + 0, w * v.x);
  atomicAdd(d + 1, w * v.y);
  atomicAdd(d + 2, w * v.z);
  atomicAdd(d + 3, w * v.w);
}

// dst[rcv[k]] += ev[k]
__global__ void scatter_sum_kernel(const float* __restrict__ ev, float* __restrict__ dst,
                                   const int* __restrict__ rcv, int E) {
  int t = blockIdx.x * blockDim.x + threadIdx.x;
  int k = t >> 5;
  if (k >= E) return;
  int c = (t & 31) << 2;
  int r = rcv[k];
  float4 v = *(const float4*)(ev + (size_t)k * H + c);
  float* d = dst + (size_t)r * H + c;
  atomicAdd(d + 0, v.x);
  atomicAdd(d + 1, v.y);
  atomicAdd(d + 2, v.z);
  atomicAdd(d + 3, v.w);
}

extern "C" void kernel_launch(void* const* d_in, const int* in_sizes, int n_in,
                              void* d_out, int out_size, void* d_ws, size_t ws_size,
                              hipStream_t stream) {
  const float* x         = (const float*)d_in[0];
  const float* edge_attr = (const float*)d_in[1];
  const int*   eidx      = (const int*)d_in[2];
  const float* W_eb_gcn  = (const float*)d_in[3];
  const float* b_eb_gcn  = (const float*)d_in[4];
  const float* W_eb_out  = (const float*)d_in[5];
  const float* b_eb_out  = (const float*)d_in[6];
  const float* W_nb_gcn  = (const float*)d_in[7];
  const float* b_nb_gcn  = (const float*)d_in[8];
  const float* W_nb_out  = (const float*)d_in[9];
  const float* b_nb_out  = (const float*)d_in[10];

  const int N = in_sizes[0] / H;
  const int E = in_sizes[1] / H;
  const int* senders   = eidx;
  const int* receivers = eidx + E;

  // workspace layout
  char*  ws    = (char*)d_ws;
  float* buf_e = (float*)ws;                          // [E,128] xw_e, later new_edge
  float* agg   = (float*)(ws + (size_t)E * H * 4);    // [N,128] agg, later h_n input
  float* xw_n  = agg + (size_t)N * H;                 // [N,128]
  float* dinv  = xw_n + (size_t)N * H;                // [N]
  int*   cnt   = (int*)(dinv + N);                    // [N]

  float* out_x = (float*)d_out;                       // [N,128]
  float* out_e = out_x + (size_t)N * H;               // [E,128] (also scratch for h_e)

  const int T = 256;

  // shared degree / dinv (identical for edge- and node-graph GCNs)
  hipMemsetAsync(cnt, 0, (size_t)N * 4, stream);
  deg_kernel<<<(E + T - 1) / T, T, 0, stream>>>(receivers, cnt, E);
  dinv_kernel<<<(N + T - 1) / T, T, 0, stream>>>(cnt, dinv, N);

  // GEMM1: xw_e = [x[s], x[r], edge_attr] @ W_eb_gcn   (K = 384)
  wmma_gemm_kernel<<<(E + TM - 1) / TM, T, 0, stream>>>(
      x, x, edge_attr, senders, receivers, W_eb_gcn,
      nullptr, nullptr, nullptr, buf_e, nullptr, E, 3 * H);

  // edge-graph GCN aggregation into out_e (used as scratch "he")
  self_scale_kernel<<<(unsigned)(((size_t)E * 32 + T - 1) / T), T, 0, stream>>>(
      buf_e, out_e, dinv, E, N);
  scatter_norm_kernel<<<(unsigned)(((size_t)E * 32 + T - 1) / T), T, 0, stream>>>(
      buf_e, out_e, senders, receivers, dinv, E);

  // GEMM2: new_edge = relu(he + b_eb_gcn) @ W_eb_out + b_eb_out
  //        out_e <- edge_attr + new_edge ; buf_e <- new_edge
  wmma_gemm_kernel<<<(E + TM - 1) / TM, T, 0, stream>>>(
      out_e, nullptr, nullptr, nullptr, nullptr, W_eb_out,
      b_eb_gcn, b_eb_out, edge_attr, out_e, buf_e, E, H);

  // agg = segment_sum(new_edge -> receivers)
  hipMemsetAsync(agg, 0, (size_t)N * H * 4, stream);
  scatter_sum_kernel<<<(unsigned)(((size_t)E * 32 + T - 1) / T), T, 0, stream>>>(
      buf_e, agg, receivers, E);

  // GEMM3: xw_n = [x, agg] @ W_nb_gcn   (K = 256)
  wmma_gemm_kernel<<<(N + TM - 1) / TM, T, 0, stream>>>(
      x, agg, nullptr, nullptr, nullptr, W_nb_gcn,
      nullptr, nullptr, nullptr, xw_n, nullptr, N, 2 * H);

  // node-graph GCN aggregation into hn (reuse agg buffer)
  self_scale_kernel<<<(unsigned)(((size_t)N * 32 + T - 1) / T), T, 0, stream>>>(
      xw_n, agg, dinv, N, N);
  scatter_norm_kernel<<<(unsigned)(((size_t)E * 32 + T - 1) / T), T, 0, stream>>>(
      xw_n, agg, senders, receivers, dinv, E);

  // GEMM4: out_x = x + relu(hn + b_nb_gcn) @ W_nb_out + b_nb_out
  wmma_gemm_kernel<<<(N + TM - 1) / TM, T, 0, stream>>>(
      agg, nullptr, nullptr, nullptr, nullptr, W_nb_out,
      b_nb_gcn, b_nb_out, x, out_x, nullptr, N, H);
}